// spatio_86079734546455
// MI455X (gfx1250) — compile-verified
//
#include <hip/hip_runtime.h>

#define BATCH 4
#define NN 2048
#define PARAM_G 0.125f
#define VIRUS_DEATH 8e-05f

typedef __attribute__((ext_vector_type(2))) float v2f;
typedef __attribute__((ext_vector_type(8))) float v8f;

// ---------------------------------------------------------------------------
// K1: col_sum[b][j] = sum_i mob[b][i][j], diagonal excluded.
// Coalesced: adjacent threads read adjacent columns j for each row i.
// ---------------------------------------------------------------------------
__global__ __launch_bounds__(256) void colsum_kernel(const float* __restrict__ mob,
                                                     float* __restrict__ cs) {
    const int b = blockIdx.x / (NN / 256);
    const int j = (blockIdx.x % (NN / 256)) * 256 + threadIdx.x;
    const float* base = mob + (size_t)b * NN * NN + j;
    float acc = 0.f;
    for (int i = 0; i < NN; ++i) {
        float v = base[(size_t)i * NN];
        acc += (i == j) ? 0.f : v;
    }
    cs[b * NN + j] = acc;
}

// ---------------------------------------------------------------------------
// K2: per batch  fai = sum(cs),  tau = sum(S+I+R),  m = fai / tau
// ---------------------------------------------------------------------------
__global__ __launch_bounds__(256) void scalars_kernel(const float* __restrict__ cs,
                                                      const float* __restrict__ SIR,
                                                      float* __restrict__ mOut) {
    __shared__ float sf[256];
    __shared__ float st[256];
    const int b = blockIdx.x;
    float fai = 0.f, tau = 0.f;
    for (int j = threadIdx.x; j < NN; j += 256) {
        fai += cs[b * NN + j];
        const float* s = SIR + ((size_t)b * NN + j) * 4;
        tau += s[0] + s[1] + s[2];
    }
    sf[threadIdx.x] = fai;
    st[threadIdx.x] = tau;
    __syncthreads();
    for (int off = 128; off > 0; off >>= 1) {
        if (threadIdx.x < off) {
            sf[threadIdx.x] += sf[threadIdx.x + off];
            st[threadIdx.x] += st[threadIdx.x + off];
        }
        __syncthreads();
    }
    if (threadIdx.x == 0) mOut[b] = sf[0] / st[0];
}

// ---------------------------------------------------------------------------
// K3 (WMMA): Y[b][i][c] = sum_j F[b][i][j] * X[b][j][c],  X = [S | I | R]
// V_WMMA_F32_16X16X4_F32:  D(16x16) = A(16x4) x B(4x16) + C
//   A layout: lane 0-15 = rows M=0..15 (K=0,1 in VGPR0,1);
//             lane 16-31 = same rows   (K=2,3)
//   B layout: VGPR0 = rows K=0 (lanes 0-15, N=lane) / K=2 (lanes 16-31);
//             VGPR1 = rows K=1 / K=3
//   D layout: element r in lane l -> M = r + 8*(l>>4), N = l&15
// Hot loop contains NO diagonal masking and NO divergent EXEC: the diagonal
// term is subtracted once after the loop, and B-columns >= 3 read a zeroed
// LDS pad slot with a stride-0 address increment.
// mob is L2-hot after K1 (67 MB < 192 MB L2).
// ---------------------------------------------------------------------------
__global__ __launch_bounds__(256) void flux_wmma_kernel(const float* __restrict__ mob,
                                                        const float* __restrict__ SIR,
                                                        float* __restrict__ y) {
    __shared__ float Xs[NN * 3 + 4];   // +4 zero pad slots for columns >= 3
    const int b = blockIdx.x / (NN / 128);
    const int rowBlk = blockIdx.x % (NN / 128);
    const int t = threadIdx.x;

    // stage S,I,R columns into LDS
    for (int idx = t; idx < NN * 3; idx += 256) {
        int j = idx / 3, comp = idx % 3;
        Xs[idx] = SIR[((size_t)b * NN + j) * 4 + comp];
    }
    if (t < 4) Xs[NN * 3 + t] = 0.f;
    __syncthreads();

    const int wave = t >> 5;
    const int lane = t & 31;
    const int m0 = rowBlk * 128 + wave * 16;
    const int M  = lane & 15;       // row within tile (A) / column of B,D
    const int kh = lane >> 4;       // which K-pair this half-wave holds
    const int row = m0 + M;
    const float* mrow = mob + ((size_t)b * NN + row) * NN;

    // B-tile LDS cursor: valid columns walk the [S,I,R] table (+12 floats per
    // k-step of 4); columns >= 3 park on the zero pad (stride 0).
    int bidx = (M < 3) ? (6 * kh + M) : (NN * 3);
    const int bstep = (M < 3) ? 12 : 0;

    v8f acc = {};
    for (int k = 0; k < NN; k += 4) {
        const int c0 = k + 2 * kh;                 // even -> 8B aligned
        const v2f a = *(const v2f*)(mrow + c0);
        v2f bb;
        bb.x = Xs[bidx];
        bb.y = Xs[bidx + 3];
        acc = __builtin_amdgcn_wmma_f32_16x16x4_f32(
            /*neg_a=*/false, a, /*neg_b=*/false, bb,
            /*c_mod=*/(short)0, acc, /*reuse_a=*/false, /*reuse_b=*/false);
        bidx += bstep;
    }

    // Columns 0..2 of D hold the S/I/R dot products; subtract the diagonal
    // contribution (F has a zeroed diagonal) and store.
    const int Ncol = lane & 15;
    const int hi = lane >> 4;
    if (Ncol < 3) {
        for (int r = 0; r < 8; ++r) {
            int m = m0 + r + 8 * hi;
            float dmob = mob[((size_t)b * NN + m) * NN + m];
            y[((size_t)b * NN + m) * 3 + Ncol] = acc[r] - dmob * Xs[m * 3 + Ncol];
        }
    }
}

// ---------------------------------------------------------------------------
// K4 (bandwidth king): arrive2[b][i][j] = { sps_m, P }  as interleaved float2,
// written as two float4 per 4 columns. One workgroup per (b, i) row.
// ---------------------------------------------------------------------------
__global__ __launch_bounds__(256) void arrive2_kernel(const float* __restrict__ mob,
                                                      const float* __restrict__ sps,
                                                      const float* __restrict__ cs,
                                                      float* __restrict__ out2) {
    const int b = blockIdx.x >> 11;          // / NN
    const int i = blockIdx.x & (NN - 1);     // % NN
    const float inv = 1.0f / cs[b * NN + i];
    const size_t rbase = ((size_t)b * NN + i) * NN;
    const float4* m4 = (const float4*)(mob + rbase);
    const float4* s4 = (const float4*)(sps + rbase);
    float4* o4 = (float4*)(out2 + rbase * 2);

    for (int q = threadIdx.x; q < NN / 4; q += 256) {
        float4 mv = m4[q];
        float4 sv = s4[q];
        const int d = i - q * 4;
        if (d >= 0 && d < 4) {               // zero the diagonal element
            ((float*)&mv)[d] = 0.f;
            ((float*)&sv)[d] = 0.f;
        }
        float4 o0, o1;
        o0.x = sv.x; o0.y = mv.x * inv; o0.z = sv.y; o0.w = mv.y * inv;
        o1.x = sv.z; o1.y = mv.z * inv; o1.z = sv.w; o1.w = mv.w * inv;
        o4[q * 2 + 0] = o0;
        o4[q * 2 + 1] = o1;
    }
}

// ---------------------------------------------------------------------------
// K5: per-(b,i) scalar math -> Ht_SIR (6) and arrive1 (4)
// ---------------------------------------------------------------------------
__global__ __launch_bounds__(256) void final_kernel(const float* __restrict__ param_b,
                                                    const float* __restrict__ contact,
                                                    const float* __restrict__ SIR,
                                                    const float* __restrict__ nb,
                                                    const float* __restrict__ nd,
                                                    const float* __restrict__ cs,
                                                    const float* __restrict__ mArr,
                                                    const float* __restrict__ y,
                                                    float* __restrict__ out) {
    const int idx = blockIdx.x * 256 + threadIdx.x;
    if (idx >= BATCH * NN) return;
    const int b = idx / NN;
    const int i = idx % NN;

    const float* s = SIR + (size_t)idx * 4;
    const float S = s[0], I = s[1], R = s[2], Isum = s[3];
    const float pb = param_b[idx];
    const float ct = contact[idx];
    const float birth = nb[i];
    const float death = nd[i];
    const float mm = mArr[b];
    const float inv = 1.0f / cs[idx];

    const float pop = S + I + R;
    const float I_new = S / pop * pb * ct * I;
    const float fS = mm * y[(size_t)idx * 3 + 0] * inv;
    const float fI = mm * y[(size_t)idx * 3 + 1] * inv;
    const float fR = mm * y[(size_t)idx * 3 + 2] * inv;

    const float R_t  = R + PARAM_G * I - death * R - mm * R + fR;
    const float I_t  = I + I_new - death * I - PARAM_G * I - VIRUS_DEATH * I - mm * I + fI;
    const float S_t  = S - I_new - death * S + birth * pop - mm * S + fS;
    const float Ist  = Isum + I_new;
    const float R0   = pb * ct / (death + PARAM_G + VIRUS_DEATH + mm);
    const float W    = pb * ct - death - PARAM_G - VIRUS_DEATH;

    float* h = out + (size_t)idx * 6;
    h[0] = R0; h[1] = I_new; h[2] = S_t; h[3] = I_t; h[4] = R_t; h[5] = Ist;

    float* a1 = out + (size_t)BATCH * NN * 6 + (size_t)idx * 4;
    a1[0] = W; a1[1] = mm; a1[2] = I; a1[3] = pop;
}

// ---------------------------------------------------------------------------
extern "C" void kernel_launch(void* const* d_in, const int* in_sizes, int n_in,
                              void* d_out, int out_size, void* d_ws, size_t ws_size,
                              hipStream_t stream) {
    const float* param_b = (const float*)d_in[0];   // (B,N,1)
    const float* contact = (const float*)d_in[1];   // (B,N,1)
    const float* mob     = (const float*)d_in[2];   // (B,N,N)
    const float* SIR     = (const float*)d_in[3];   // (B,N,4)
    const float* sps     = (const float*)d_in[4];   // (B,N,N)
    const float* nb      = (const float*)d_in[5];   // (N,1)
    const float* nd      = (const float*)d_in[6];   // (N,1)

    float* out = (float*)d_out;
    // output layout: Ht_SIR [B*N*6] | arrive1 [B*N*4] | arrive2 [B*N*N*2]
    float* out2 = out + (size_t)BATCH * NN * 6 + (size_t)BATCH * NN * 4;

    float* ws   = (float*)d_ws;
    float* cs   = ws;                         // B*N
    float* mArr = ws + BATCH * NN;            // B
    float* y    = ws + BATCH * NN + 16;       // B*N*3

    colsum_kernel   <<<BATCH * (NN / 256), 256, 0, stream>>>(mob, cs);
    scalars_kernel  <<<BATCH,              256, 0, stream>>>(cs, SIR, mArr);
    flux_wmma_kernel<<<BATCH * (NN / 128), 256, 0, stream>>>(mob, SIR, y);
    arrive2_kernel  <<<BATCH * NN,         256, 0, stream>>>(mob, sps, cs, out2);
    final_kernel    <<<(BATCH * NN + 255) / 256, 256, 0, stream>>>(
        param_b, contact, SIR, nb, nd, cs, mArr, y, out);
}